// YOLOLayer_74586402062689
// MI455X (gfx1250) — compile-verified
//
#include <hip/hip_runtime.h>

// ---------------- problem constants (from reference) ----------------
#define NUM_ANCHORS 3
#define ATTRS       85            // 5 + 80 classes
#define GD          44            // grid H = W
#define CELLS       (GD * GD)     // 1936
#define STRIDE_F    8.0f          // 352/44
#define TILE_ROWS   4
#define TILE_CELLS  (TILE_ROWS * GD)       // 176 cells per block
#define NTILES      (GD / TILE_ROWS)       // 11 tiles per (b,anchor)
#define TILE_FLOATS (ATTRS * TILE_CELLS)   // 14960 floats (= in = out per tile)
#define TILE_VEC4   (TILE_FLOATS / 4)      // 3740 float4 transfers
#define LOG2E       1.4426950408889634f

typedef int v4i __attribute__((vector_size(16)));   // matches builtin's pointee type

__device__ __forceinline__ float fast_sigmoid(float x) {
    // 1 / (1 + exp(-x)) via v_exp_f32 + v_rcp_f32 (TRANS ops, co-execute with VALU)
    return __builtin_amdgcn_rcpf(1.0f + __builtin_amdgcn_exp2f(-x * LOG2E));
}
__device__ __forceinline__ float fast_exp(float x) {
    return __builtin_amdgcn_exp2f(x * LOG2E);
}

__global__ __launch_bounds__(256)
void yolo_head_kernel(const float* __restrict__ inp,      // [BS, 255, 44, 44]
                      const float* __restrict__ sanchors, // [3, 2] (already / stride)
                      float* __restrict__ out) {          // [BS, 3*1936, 85]
    // One output tile per block: 176 consecutive cells of one (b, anchor) pair.
    __shared__ __align__(16) float lds[TILE_FLOATS];

    const int tile = blockIdx.x;          // 0 .. 10  (4 grid rows each)
    const int ba   = blockIdx.y;          // 0 .. BS*3-1 : (b*3 + anchor)
    const int a    = ba % NUM_ANCHORS;
    const int tid  = threadIdx.x;

    // anchor * stride folded in: out_wh = exp(p) * scaled_anchor * stride
    const float aw = sanchors[a * 2 + 0] * STRIDE_F;
    const float ah = sanchors[a * 2 + 1] * STRIDE_F;

    // ---- phase 1: coalesced float4 loads, transform, transpose into LDS ----
    // input float offset of this tile: channel-major planes of 1936 floats
    const long long in_base = (long long)ba * (ATTRS * CELLS) + (long long)tile * TILE_CELLS;

    for (int v = tid; v < TILE_VEC4; v += 256) {
        const int attr = v / GD;            // 0..84
        const int cq   = v - attr * GD;     // 0..43 (float4 group within the 176-cell row)
        const int lc0  = cq * 4;            // local cell 0..172
        const float4 p = *(const float4*)(inp + in_base + (long long)attr * CELLS + lc0);
        float r[4] = {p.x, p.y, p.z, p.w};
#pragma unroll
        for (int j = 0; j < 4; ++j) {
            const int lc = lc0 + j;         // local cell 0..175
            float val;
            if (attr == 0) {
                const int gx = lc % GD;     // tile*176 is a multiple of 44
                val = (fast_sigmoid(r[j]) + (float)gx) * STRIDE_F;
            } else if (attr == 1) {
                const int gy = tile * TILE_ROWS + lc / GD;
                val = (fast_sigmoid(r[j]) + (float)gy) * STRIDE_F;
            } else if (attr == 2) {
                val = fast_exp(r[j]) * aw;
            } else if (attr == 3) {
                val = fast_exp(r[j]) * ah;
            } else {
                val = fast_sigmoid(r[j]);   // conf + 80 classes
            }
            lds[lc * ATTRS + attr] = val;   // transpose: attr becomes fastest dim
        }
    }
    __syncthreads();

    // ---- phase 2: LDS now holds the EXACT contiguous output byte stream ----
    // out float offset: ((ba*1936 + tile*176) * 85), multiple of 4 -> 16B aligned
    const long long out_base = ((long long)ba * CELLS + (long long)tile * TILE_CELLS) * (long long)ATTRS;
    float* outp = out + out_base;

#if defined(__gfx1250__)
    {
        __attribute__((address_space(1))) v4i* gdst =
            (__attribute__((address_space(1))) v4i*)outp;
        __attribute__((address_space(3))) v4i* lsrc =
            (__attribute__((address_space(3))) v4i*)lds;
        for (int f = tid; f < TILE_VEC4; f += 256) {
#if __has_builtin(__builtin_amdgcn_global_store_async_from_lds_b128)
            __builtin_amdgcn_global_store_async_from_lds_b128(
                gdst + f, lsrc + f, /*offset=*/0, /*cpol=*/0);
#else
            unsigned long long gaddr = (unsigned long long)(size_t)(gdst + f);
            unsigned           laddr = (unsigned)(size_t)(lsrc + f);
            asm volatile("global_store_async_from_lds_b128 %0, %1, off"
                         :: "v"(gaddr), "v"(laddr) : "memory");
#endif
        }
#if __has_builtin(__builtin_amdgcn_s_wait_asynccnt)
        __builtin_amdgcn_s_wait_asynccnt(0);
#else
        asm volatile("s_wait_asynccnt 0" ::: "memory");
#endif
    }
#else
    // Host-parse / non-gfx1250 fallback: plain coalesced float4 copy.
    for (int f = tid; f < TILE_VEC4; f += 256) {
        float4 v = *(float4*)&lds[f * 4];
        *(float4*)(outp + (long long)f * 4) = v;
    }
#endif
}

extern "C" void kernel_launch(void* const* d_in, const int* in_sizes, int n_in,
                              void* d_out, int out_size, void* d_ws, size_t ws_size,
                              hipStream_t stream) {
    const float* inp = (const float*)d_in[0];       // [BS,255,44,44] f32
    const float* sa  = (const float*)d_in[1];       // [3,2] f32 scaled anchors
    float* out       = (float*)d_out;               // [BS,5808,85] f32

    const int ba_count = in_sizes[0] / (ATTRS * CELLS); // BS * 3 = 192
    dim3 grid(NTILES, ba_count);                         // 11 x 192 blocks
    yolo_head_kernel<<<grid, 256, 0, stream>>>(inp, sa, out);
}